// LocalContrastNormalization_45792941310388
// MI455X (gfx1250) — compile-verified
//
#include <hip/hip_runtime.h>
#include <math.h>
#include <stdint.h>

typedef float v2f __attribute__((ext_vector_type(2)));
typedef float v8f __attribute__((ext_vector_type(8)));

#define HIMG 512
#define WIMG 512
#define NBATCH 16
#define NPIX (NBATCH * HIMG * WIMG)   // 4,194,304 pixels
#define PADW 520                      // 512 + 8 halo
#define ROWS 4                        // output rows per block
#define HALO (ROWS + 8)               // 12 input rows per block

// ---------------------------------------------------------------------------
// Kernel 1: normalize filter  f = (filt + 10) / sum(filt + 10)
// ---------------------------------------------------------------------------
__global__ void prep_filter(const float* __restrict__ filt, float* __restrict__ f) {
    __shared__ float sf[81];
    __shared__ float ssum;
    int t = threadIdx.x;
    if (t < 81) sf[t] = filt[t] + 10.0f;
    __syncthreads();
    if (t == 0) {
        float s = 0.f;
        for (int i = 0; i < 81; ++i) s += sf[i];
        ssum = s;
    }
    __syncthreads();
    if (t < 81) f[t] = sf[t] / ssum;
}

// ---------------------------------------------------------------------------
// Kernel 2/3: 9x9 SAME conv of a per-pixel scalar field via WMMA f32 16x16x4.
// Block = 4 image rows (b, y0..y0+3), 256 threads = 8 waves.
// Wave w: output row y0 + (w>>1), pixel segment x0 = (w&1)*256.
//
// PASS2==0: field = sum_c x[c]           -> dst = local mean
// PASS2==1: field = sum_c (x[c]-mean)^2  -> dst = sigma = sqrt(conv), + row sums
//           (lm halo tile staged via global_load_async_to_lds)
//
// Toeplitz GEMM per (row, dy): D[m,n] = sum_k A[m,k] * B[k,n],
//   A[m,k] = field[row+dy-4][x0-4+16m+k]       (K = 24, chunks of 4)
//   B[k,n] = f[dy][k-n] if 0<=k-n<9 else 0     (pre-paired in LDS table)
// ---------------------------------------------------------------------------
template <int PASS2>
__global__ __launch_bounds__(256) void conv9_wmma(
    const float* __restrict__ x,     // (16,512,512,3) NHWC
    const float* __restrict__ lm,    // local mean (PASS2 only)
    const float* __restrict__ f,     // 81 normalized weights
    float* __restrict__ dst,         // per-pixel scalar result
    float* __restrict__ partial)     // per-row sigma sums (PASS2 only)
{
    __shared__ __align__(16) float srows[HALO][PADW];       // conv input field
    __shared__ __align__(8)  float pt[9 * 40 * 2];          // paired Toeplitz filter
    __shared__ float red[256];
    __shared__ __align__(16) float lmbuf[PASS2 ? HALO * PADW : 4];

    const int tid  = threadIdx.x;
    const int blk  = blockIdx.x;           // b*128 + y0/4
    const int b    = blk >> 7;
    const int y0   = (blk & 127) << 2;

    // gfx1250 prefetch of rows the next block will need
    {
        int pry = y0 + 8; if (pry > 511) pry = 511;
        const float* pf = x + (((size_t)((b << 9) | pry) << 9) + (size_t)(tid << 1)) * 3;
        __builtin_prefetch(pf, 0, 1);
    }

    // Paired zero-padded Toeplitz filter table: pt[dy][i] = {fpad[i-16], fpad[i-15]}
    for (int i = tid; i < 9 * 40; i += 256) {
        const int dy = i / 40;
        const int d  = (i - dy * 40) - 16;
        pt[2 * i]     = ((unsigned)d       < 9u) ? f[dy * 9 + d]     : 0.f;
        pt[2 * i + 1] = ((unsigned)(d + 1) < 9u) ? f[dy * 9 + d + 1] : 0.f;
    }

    if (PASS2) {
        // Async-stage the local-mean halo tile into LDS (ASYNCcnt-tracked DMA path)
        for (int i = tid; i < HALO * PADW; i += 256) {
            const int dyr = i / PADW;
            const int col = i - dyr * PADW;
            const int ry  = y0 + dyr - 4;
            const int rx  = col - 4;
            if ((unsigned)ry < 512u && (unsigned)rx < 512u) {
                const uint64_t gp = (uint64_t)(uintptr_t)(lm + ((((size_t)((b << 9) | ry)) << 9) + rx));
                const uint32_t lp = (uint32_t)(uintptr_t)(&lmbuf[i]);
                asm volatile("global_load_async_to_lds_b32 %0, %1, off"
                             :: "v"(lp), "v"(gp) : "memory");
            } else {
                lmbuf[i] = 0.f;
            }
        }
        asm volatile("s_wait_asynccnt 0" ::: "memory");
        __syncthreads();
    }

    // Build conv input field (halo tile, zero-padded SAME boundary)
    for (int i = tid; i < HALO * PADW; i += 256) {
        const int dyr = i / PADW;
        const int col = i - dyr * PADW;
        const int ry  = y0 + dyr - 4;
        const int rx  = col - 4;
        float val = 0.f;
        if ((unsigned)ry < 512u && (unsigned)rx < 512u) {
            const size_t pix = (((size_t)((b << 9) | ry)) << 9) + (size_t)rx;
            const float* px = x + pix * 3;
            if (PASS2) {
                const float mm = lmbuf[i];
                const float d0 = px[0] - mm, d1 = px[1] - mm, d2 = px[2] - mm;
                val = d0 * d0 + d1 * d1 + d2 * d2;
            } else {
                val = px[0] + px[1] + px[2];
            }
        }
        srows[dyr][col] = val;
    }
    __syncthreads();

    const int lane = tid & 31;
    const int wave = tid >> 5;
    const int rblk = wave >> 1;          // row within block (0..3)
    const int x0   = (wave & 1) << 8;    // 0 or 256: pixel segment
    const int n    = lane & 15;          // N index (output column)
    const int hk   = lane >> 4;          // which K half this lane holds
    const int mrow = lane & 15;          // M index for A operand

    v8f acc = {0.f, 0.f, 0.f, 0.f, 0.f, 0.f, 0.f, 0.f};

    const int abase = x0 + 16 * mrow + 2 * hk;       // even -> 8B aligned
    const int bbase = 16 + 2 * hk - n;               // in [1..18]

    for (int dy = 0; dy < 9; ++dy) {
        const float* srow = &srows[rblk + dy][0];    // srow[4+x] == field at col x
        const float* prow = &pt[2 * (dy * 40 + bbase)];
#pragma unroll
        for (int kk = 0; kk < 6; ++kk) {
            v2f a  = *(const v2f*)(srow + abase + 4 * kk);
            v2f bv = *(const v2f*)(prow + 8 * kk);
            acc = __builtin_amdgcn_wmma_f32_16x16x4_f32(
                false, a, false, bv, (short)0, acc, false, false);
        }
    }

    // Scatter D (C/D layout: vgpr r -> M = r + 8*hk, N = n), optional sqrt+sum.
    const size_t rowbase = ((size_t)(blk * 4 + rblk)) << 9;
    float lsum = 0.f;
#pragma unroll
    for (int r = 0; r < 8; ++r) {
        const int mm = r + 8 * hk;
        const int px = x0 + 16 * mm + n;
        float v = acc[r];
        if (PASS2) {
            v = (v > 0.f) ? sqrtf(v) : 0.f;
            lsum += v;
        }
        dst[rowbase + (size_t)px] = v;
    }

    if (PASS2) {
        red[tid] = lsum;
        __syncthreads();
        // Reduce within each 64-thread group (= one output row)
        for (int s = 32; s > 0; s >>= 1) {
            if ((tid & 63) < s) red[tid] += red[tid + s];
            __syncthreads();
        }
        if (tid < ROWS) partial[blk * 4 + tid] = red[tid * 64];
    }
}

// ---------------------------------------------------------------------------
// Kernel 4: per-batch mean of sigma from 512 per-row partial sums.
// ---------------------------------------------------------------------------
__global__ void reduce_mean(const float* __restrict__ partial, float* __restrict__ mean) {
    __shared__ float red[256];
    const int b = blockIdx.x, t = threadIdx.x;
    red[t] = partial[b * 512 + t] + partial[b * 512 + 256 + t];
    __syncthreads();
    for (int s = 128; s > 0; s >>= 1) {
        if (t < s) red[t] += red[t + s];
        __syncthreads();
    }
    if (t == 0) mean[b] = red[0] * (1.0f / (512.0f * 512.0f));
}

// ---------------------------------------------------------------------------
// Kernel 5: out = (x - local_mean) / max(mean_b, sigma)
// ---------------------------------------------------------------------------
__global__ __launch_bounds__(256) void finalize(
    const float* __restrict__ x, const float* __restrict__ lm,
    const float* __restrict__ sigma, const float* __restrict__ mean,
    float* __restrict__ out)
{
    const size_t p = (size_t)blockIdx.x * 256 + threadIdx.x;   // pixel index
    const int b = (int)(p >> 18);                              // 512*512 = 2^18
    const float mm = lm[p];
    const float sg = sigma[p];
    const float mx = fmaxf(mean[b], sg);
    const float* px = x + p * 3;
    float* po = out + p * 3;
    po[0] = (px[0] - mm) / mx;
    po[1] = (px[1] - mm) / mx;
    po[2] = (px[2] - mm) / mx;
}

// ---------------------------------------------------------------------------
extern "C" void kernel_launch(void* const* d_in, const int* in_sizes, int n_in,
                              void* d_out, int out_size, void* d_ws, size_t ws_size,
                              hipStream_t stream) {
    const float* x    = (const float*)d_in[0];   // (16,512,512,3) fp32
    const float* filt = (const float*)d_in[1];   // (9,9,1,1) fp32
    float* out = (float*)d_out;

    float* ws      = (float*)d_ws;
    float* f       = ws;                  // 128 floats (81 used)
    float* lmean   = f + 128;             // NPIX
    float* sigma   = lmean + NPIX;        // NPIX
    float* partial = sigma + NPIX;        // 16*512 row sums
    float* meanb   = partial + NBATCH * HIMG;  // 16

    prep_filter<<<1, 96, 0, stream>>>(filt, f);
    conv9_wmma<0><<<NBATCH * (HIMG / ROWS), 256, 0, stream>>>(x, nullptr, f, lmean, nullptr);
    conv9_wmma<1><<<NBATCH * (HIMG / ROWS), 256, 0, stream>>>(x, lmean, f, sigma, partial);
    reduce_mean<<<NBATCH, 256, 0, stream>>>(partial, meanb);
    finalize<<<NPIX / 256, 256, 0, stream>>>(x, lmean, sigma, meanb, out);
}